// IEMC_30906584662138
// MI455X (gfx1250) — compile-verified
//
#include <hip/hip_runtime.h>
#include <math.h>

// ---------------------------------------------------------------------------
// IEMC contrastive loss, MI455X (gfx1250, wave32).
// Memory-bound: 3 x [8192x1024] f32 = 96MB streamed once -> ~4.1us floor at
// 23.3 TB/s. Single fused pass per matrix: row-norm (VALU pk_fma) + two
// anchor dots (v_wmma_f32_16x16x4_f32, A = 16x4 row tile, B cols 0/1 =
// anchors, other cols read a zero LDS column so the loop is branch-free).
// ---------------------------------------------------------------------------

typedef __attribute__((ext_vector_type(2))) float v2f;
typedef __attribute__((ext_vector_type(8))) float v8f;

#define BQ 8192
#define DD 1024
#define NCLS 7
#define NBLK 128            // main-kernel grid
#define ROWS_PER_BLOCK 64   // 4 waves * 16 rows

// Workspace layout (d_ws):
//   int   ws[0] = anchor index
//   int   ws[1] = a_cls
//   int   ws[2] = pos_num
//   float ws[3] = 1/||anc_t||,  ws[4] = 1/||anc_v||,  ws[5] = 1/||anc_a||
//   float ws[8 .. 8+NBLK)        = per-block pos partial sums
//   float ws[8+NBLK .. 8+2NBLK)  = per-block neg partial sums

__global__ __launch_bounds__(256)
void imc_prep(const int* __restrict__ label,
              const float* __restrict__ s_t, const float* __restrict__ s_v,
              const float* __restrict__ s_a,
              int* __restrict__ wsI, float* __restrict__ wsF)
{
    __shared__ int cnt[NCLS];
    __shared__ int sAnchor;
    __shared__ int sCls;
    __shared__ float red0[256], red1[256], red2[256];

    const int tid = threadIdx.x;
    if (tid < NCLS) cnt[tid] = 0;
    if (tid == 0) sAnchor = BQ;
    __syncthreads();

    for (int i = tid; i < BQ; i += 256) atomicAdd(&cnt[label[i]], 1);
    __syncthreads();

    if (tid == 0) {
        int c = 0;
        for (int k = 0; k < NCLS; ++k) { if (cnt[k] > 1) { c = k; break; } }
        sCls = c;
    }
    __syncthreads();

    const int acls = sCls;
    for (int i = tid; i < BQ; i += 256)
        if (label[i] == acls) atomicMin(&sAnchor, i);
    __syncthreads();

    int anchor = sAnchor;
    if (anchor >= BQ) anchor = 0;   // degenerate fallback (matches argmax-of-all-false)

    // anchor vector squared norms
    float t = 0.f, v = 0.f, a = 0.f;
    const size_t ab = (size_t)anchor * DD;
    for (int i = tid; i < DD; i += 256) {
        float x = s_t[ab + i]; t += x * x;
        float y = s_v[ab + i]; v += y * y;
        float z = s_a[ab + i]; a += z * z;
    }
    red0[tid] = t; red1[tid] = v; red2[tid] = a;
    __syncthreads();
    for (int off = 128; off > 0; off >>= 1) {
        if (tid < off) {
            red0[tid] += red0[tid + off];
            red1[tid] += red1[tid + off];
            red2[tid] += red2[tid + off];
        }
        __syncthreads();
    }
    if (tid == 0) {
        wsI[0] = anchor;
        wsI[1] = acls;
        wsI[2] = cnt[acls];
        wsF[3] = 1.0f / sqrtf(red0[0]);
        wsF[4] = 1.0f / sqrtf(red1[0]);
        wsF[5] = 1.0f / sqrtf(red2[0]);
    }
}

__global__ __launch_bounds__(128)
void imc_main(const int* __restrict__ label,
              const float* __restrict__ s_t, const float* __restrict__ s_v,
              const float* __restrict__ s_a,
              const int* __restrict__ wsI, float* __restrict__ wsF)
{
    // sB[0]=anc_t, sB[1]=anc_v, sB[2]=anc_a, sB[3]=zeros (for B-frag lanes N>=2)
    __shared__ __align__(16) float sB[4][DD];
    __shared__ float sDot[3][4][16][2];   // [matrix][wave][row][which-anchor]
    __shared__ float sNorm[3][4][16];
    __shared__ float sPos[4], sNeg[4];

    const int tid  = threadIdx.x;
    const int lane = tid & 31;
    const int wave = tid >> 5;
    const int anchor = wsI[0];
    const int acls   = wsI[1];

    // Stage anchors + zero column in LDS (16KB): anchors hit HBM/L2 once per block.
    {
        const size_t ab = (size_t)anchor * DD;
        for (int i = tid; i < DD; i += 128) {
            sB[0][i] = s_t[ab + i];
            sB[1][i] = s_v[ab + i];
            sB[2][i] = s_a[ab + i];
            sB[3][i] = 0.0f;
        }
    }
    __syncthreads();

    const int rowBase = blockIdx.x * ROWS_PER_BLOCK + wave * 16;
    const int rlane = lane & 15;           // A-frag: matrix row within tile
    const int khalf = (lane >> 4) << 1;    // A/B frag: K sub-offset (0 or 2)
    const int nId   = lane & 15;           // B/D frag: column index

    const float* rowT = s_t + (size_t)(rowBase + rlane) * DD;
    const float* rowV = s_v + (size_t)(rowBase + rlane) * DD;
    const float* rowA = s_a + (size_t)(rowBase + rlane) * DD;

    // Loop-invariant per-lane B-fragment sources (branch-free inner loop):
    //   s_t dots vs (anc_a, anc_v); s_v dots vs (anc_t, anc_a); s_a dots vs (anc_t, anc_v)
    const float* bT = (nId == 0) ? sB[2] : (nId == 1) ? sB[1] : sB[3];
    const float* bV = (nId == 0) ? sB[0] : (nId == 1) ? sB[2] : sB[3];
    const float* bA = (nId == 0) ? sB[0] : (nId == 1) ? sB[1] : sB[3];

    v8f accT = {}; v8f accV = {}; v8f accA = {};
    float nrmT = 0.f, nrmV = 0.f, nrmA = 0.f;

    #pragma unroll 4
    for (int kc = 0; kc < DD; kc += 4) {
        const int ki = kc + khalf;
        // A fragments: 16x4 fp32 tile (2 regs/lane), straight from global (L0-resident lines)
        v2f at = *(const v2f*)(rowT + ki);
        v2f av = *(const v2f*)(rowV + ki);
        v2f aa = *(const v2f*)(rowA + ki);
        // B fragments: unconditional ds_load_b64 (zero column for lanes N>=2)
        v2f fbT = *(const v2f*)(bT + ki);
        v2f fbV = *(const v2f*)(bV + ki);
        v2f fbA = *(const v2f*)(bA + ki);
        // row norms ride along in (dual-issue / packed) VALU
        nrmT = __builtin_fmaf(at.x, at.x, nrmT); nrmT = __builtin_fmaf(at.y, at.y, nrmT);
        nrmV = __builtin_fmaf(av.x, av.x, nrmV); nrmV = __builtin_fmaf(av.y, av.y, nrmV);
        nrmA = __builtin_fmaf(aa.x, aa.x, nrmA); nrmA = __builtin_fmaf(aa.y, aa.y, nrmA);
        accT = __builtin_amdgcn_wmma_f32_16x16x4_f32(false, at, false, fbT, (short)0, accT, false, false);
        accV = __builtin_amdgcn_wmma_f32_16x16x4_f32(false, av, false, fbV, (short)0, accV, false, false);
        accA = __builtin_amdgcn_wmma_f32_16x16x4_f32(false, aa, false, fbA, (short)0, accA, false, false);
    }

    // Unscramble D layout (VGPR j <-> M=j / j+8, lane <-> N) through LDS.
    const int mrow = (lane >> 4) * 8;
    if (nId < 2) {
        #pragma unroll
        for (int j = 0; j < 8; ++j) {
            sDot[0][wave][mrow + j][nId] = accT[j];
            sDot[1][wave][mrow + j][nId] = accV[j];
            sDot[2][wave][mrow + j][nId] = accA[j];
        }
    }
    // Combine the two K-halves of the per-row norms.
    float fT = nrmT + __shfl_xor(nrmT, 16, 32);
    float fV = nrmV + __shfl_xor(nrmV, 16, 32);
    float fA = nrmA + __shfl_xor(nrmA, 16, 32);
    if (lane < 16) {
        sNorm[0][wave][lane] = fT;
        sNorm[1][wave][lane] = fV;
        sNorm[2][wave][lane] = fA;
    }
    __syncthreads();

    float pc = 0.f, nc = 0.f;
    if (lane < 16) {
        const int row = rowBase + lane;
        const float invT = 1.0f / sqrtf(sNorm[0][wave][lane]);
        const float invV = 1.0f / sqrtf(sNorm[1][wave][lane]);
        const float invA = 1.0f / sqrtf(sNorm[2][wave][lane]);
        const float iAT = wsF[3], iAV = wsF[4], iAA = wsF[5];
        float C = (sDot[0][wave][lane][0] * iAA + sDot[0][wave][lane][1] * iAV) * invT
                + (sDot[1][wave][lane][0] * iAT + sDot[1][wave][lane][1] * iAA) * invV
                + (sDot[2][wave][lane][0] * iAT + sDot[2][wave][lane][1] * iAV) * invA;
        if (row != anchor) {
            if (label[row] == acls) pc = C; else nc = C;
        }
    }
    #pragma unroll
    for (int off = 16; off > 0; off >>= 1) {
        pc += __shfl_xor(pc, off, 32);
        nc += __shfl_xor(nc, off, 32);
    }
    if (lane == 0) { sPos[wave] = pc; sNeg[wave] = nc; }
    __syncthreads();
    if (tid == 0) {
        wsF[8 + blockIdx.x]        = sPos[0] + sPos[1] + sPos[2] + sPos[3];
        wsF[8 + NBLK + blockIdx.x] = sNeg[0] + sNeg[1] + sNeg[2] + sNeg[3];
    }
}

__global__ void imc_final(const int* __restrict__ wsI, const float* __restrict__ wsF,
                          float* __restrict__ out)
{
    if (threadIdx.x == 0 && blockIdx.x == 0) {
        float pos = 0.f, neg = 0.f;
        for (int i = 0; i < NBLK; ++i) {      // fixed order -> deterministic
            pos += wsF[8 + i];
            neg += wsF[8 + NBLK + i];
        }
        const float pn = (float)wsI[2];
        const float nn = (float)BQ - pn;
        out[0] = (6.0f - pos / pn + neg / nn) / 3.0f;
    }
}

extern "C" void kernel_launch(void* const* d_in, const int* in_sizes, int n_in,
                              void* d_out, int out_size, void* d_ws, size_t ws_size,
                              hipStream_t stream) {
    const int*   label = (const int*)d_in[0];
    const float* s_t   = (const float*)d_in[1];
    const float* s_v   = (const float*)d_in[2];
    const float* s_a   = (const float*)d_in[3];
    float* out = (float*)d_out;
    int*   wsI = (int*)d_ws;
    float* wsF = (float*)d_ws;
    (void)in_sizes; (void)n_in; (void)out_size; (void)ws_size;

    imc_prep <<<1,    256, 0, stream>>>(label, s_t, s_v, s_a, wsI, wsF);
    imc_main <<<NBLK, 128, 0, stream>>>(label, s_t, s_v, s_a, wsI, wsF);
    imc_final<<<1,     32, 0, stream>>>(wsI, wsF, out);
}